// MemoryUpdate_46780783788618
// MI455X (gfx1250) — compile-verified
//
#include <hip/hip_runtime.h>
#include <hip/hip_bf16.h>
#include <math.h>

// MemoryUpdate for MI455X (gfx1250, wave32).
//
// Algebraic collapse: energy[b,m] = (query W_q W_k^T W_trans^T)[b] . emb[tok[b,m]] / sqrt(D)
//   -> 3 tiny 32x256x256 GEMMs (WMMA f32 16x16x4) + L2-resident gather-dot.
// v = x @ W_v is dead code in the reference and is skipped.
// emb_table is 50000*256*4 = 51.2 MB -> fully resident in the 192 MB L2;
// the 268 MB of gathered row reads run at L2 bandwidth, not HBM.
// Top-k(512 of 8192) per batch: full bitonic sort of (value,index) in LDS
// (32 KB f32 values + 16 KB u16 indices = 48 KB of the WGP's 320 KB),
// descending value with ascending-index tiebreak == jax.lax.top_k order.

#define BS       32
#define MEM_LEN  8192
#define D_MODEL  256
#define MEM_SIZE 512

typedef __attribute__((ext_vector_type(2))) float v2f;
typedef __attribute__((ext_vector_type(8))) float v8f;

// ---------------------------------------------------------------------------
// Stage 1: one step of the q_eff chain: C[32x256] = A[32x256] * (B or B^T).
// TRANSB is a compile-time template arg so the K-loop body specializes to
// two global_load_b64 + one v_wmma with no per-iteration branching.
// One workgroup, 16 waves; wave w owns N-tile w (16 cols), loops over 2 M-tiles.
// A frag (16x4):  lanes 0-15 -> K={0,1}, lanes 16-31 -> K={2,3}, M = lane&15.
// B frag (4x16):  lanes 0-15 -> K={0,1}, lanes 16-31 -> K={2,3}, N = lane&15.
// C frag (16x16): VGPR r -> M = r + 8*(lane>>4), N = lane&15.
// ---------------------------------------------------------------------------
template <int TRANSB>
__global__ __launch_bounds__(512) void gemm_qchain(const float* __restrict__ A,
                                                   const float* __restrict__ B,
                                                   float* __restrict__ C) {
  const int lane = threadIdx.x & 31;
  const int wave = threadIdx.x >> 5;   // 0..15 -> N tile
  const int n0   = wave * 16;
  const int half = lane >> 4;          // 0 or 1
  const int l15  = lane & 15;

  for (int mt = 0; mt < 2; ++mt) {
    const int m0 = mt * 16;
    v8f acc = {};
#pragma unroll 4
    for (int kb = 0; kb < D_MODEL / 4; ++kb) {
      const int k = kb * 4 + half * 2;
      v2f a = *(const v2f*)(A + (m0 + l15) * D_MODEL + k);
      v2f bfrag;
      if (TRANSB) {            // B_eff[k][n] = B[n][k]  (contiguous pair in k)
        bfrag = *(const v2f*)(B + (n0 + l15) * D_MODEL + k);
      } else {                 // B_eff[k][n] = B[k][n]
        bfrag.x = B[(k + 0) * D_MODEL + n0 + l15];
        bfrag.y = B[(k + 1) * D_MODEL + n0 + l15];
      }
      acc = __builtin_amdgcn_wmma_f32_16x16x4_f32(
          /*neg_a=*/false, a, /*neg_b=*/false, bfrag,
          /*c_mod=*/(short)0, acc, /*reuse_a=*/false, /*reuse_b=*/false);
    }
#pragma unroll
    for (int r = 0; r < 8; ++r) {
      const int m = m0 + r + half * 8;
      C[m * D_MODEL + n0 + l15] = acc[r];
    }
  }
}

// ---------------------------------------------------------------------------
// Stage 2: energy[b,m] = q_eff[b] . emb[tok[b,m]] / 16.
// One wave per (b,m): each lane owns 8 consecutive floats of the 256-dim row
// (two float4 loads, fully coalesced across the wave), wave32 xor-reduce.
// ---------------------------------------------------------------------------
__global__ __launch_bounds__(256) void energy_kernel(const int*   __restrict__ mem_input,
                                                     const float* __restrict__ qeff,
                                                     const float* __restrict__ emb,
                                                     float*       __restrict__ energy) {
  const int b    = blockIdx.y;
  const int lane = threadIdx.x & 31;
  const int wave = threadIdx.x >> 5;              // 0..7
  const int m    = blockIdx.x * 8 + wave;

  const float4 q0 = *(const float4*)(qeff + b * D_MODEL + lane * 8);
  const float4 q1 = *(const float4*)(qeff + b * D_MODEL + lane * 8 + 4);

  const int tok   = mem_input[b * MEM_LEN + m];   // broadcast load within wave
  const float* e  = emb + (long)tok * D_MODEL + lane * 8;
  const float4 e0 = *(const float4*)(e);
  const float4 e1 = *(const float4*)(e + 4);

  float s = q0.x * e0.x + q0.y * e0.y + q0.z * e0.z + q0.w * e0.w
          + q1.x * e1.x + q1.y * e1.y + q1.z * e1.z + q1.w * e1.w;
#pragma unroll
  for (int off = 16; off > 0; off >>= 1) s += __shfl_xor(s, off, 32);
  if (lane == 0) energy[b * MEM_LEN + m] = s * (1.0f / 16.0f);  // / sqrt(256)
}

// ---------------------------------------------------------------------------
// Stage 3: per-batch softmax over 8192, then mask-zeroing AFTER softmax
// (matches reference). One block per batch, 1024 threads x 8 elements.
// ---------------------------------------------------------------------------
__global__ __launch_bounds__(1024) void softmax_mask_kernel(const float* __restrict__ energy,
                                                            const int*   __restrict__ mask,
                                                            float*       __restrict__ attn) {
  const int b = blockIdx.x;
  const int t = threadIdx.x;
  __shared__ float red[1024];

  const float* e = energy + b * MEM_LEN;
  float loc[8];
  float mx = -3.4e38f;
#pragma unroll
  for (int j = 0; j < 8; ++j) {
    loc[j] = e[j * 1024 + t];
    mx = fmaxf(mx, loc[j]);
  }
  red[t] = mx;
  __syncthreads();
  for (int s2 = 512; s2 > 0; s2 >>= 1) {
    if (t < s2) red[t] = fmaxf(red[t], red[t + s2]);
    __syncthreads();
  }
  mx = red[0];
  __syncthreads();

  float sum = 0.0f;
#pragma unroll
  for (int j = 0; j < 8; ++j) {
    loc[j] = __expf(loc[j] - mx);
    sum += loc[j];
  }
  red[t] = sum;
  __syncthreads();
  for (int s2 = 512; s2 > 0; s2 >>= 1) {
    if (t < s2) red[t] += red[t + s2];
    __syncthreads();
  }
  const float inv = 1.0f / red[0];

#pragma unroll
  for (int j = 0; j < 8; ++j) {
    const int m = j * 1024 + t;
    float a = loc[j] * inv;
    if (mask[b * MEM_LEN + m] == 0) a = 0.0f;
    attn[b * MEM_LEN + m] = a;
  }
}

// ---------------------------------------------------------------------------
// Stage 4: per-batch top-512 via full in-LDS bitonic sort of 8192 keys
// (descending value, ascending index on ties -> jax.lax.top_k order),
// then gather mem_input/mask at the winning indices. 48 KB LDS / block.
// ---------------------------------------------------------------------------
__global__ __launch_bounds__(1024) void topk_gather_kernel(const float* __restrict__ attn,
                                                           const int*   __restrict__ mem_input,
                                                           const int*   __restrict__ mask,
                                                           int*         __restrict__ mem_out,
                                                           int*         __restrict__ mask_out) {
  const int b = blockIdx.x;
  const int t = threadIdx.x;
  __shared__ float          sv[MEM_LEN];
  __shared__ unsigned short si[MEM_LEN];

  for (int i = t; i < MEM_LEN; i += 1024) {
    sv[i] = attn[b * MEM_LEN + i];
    si[i] = (unsigned short)i;
  }
  __syncthreads();

  for (int k = 2; k <= MEM_LEN; k <<= 1) {
    for (int j = k >> 1; j > 0; j >>= 1) {
      for (int i = t; i < MEM_LEN; i += 1024) {
        const int ixj = i ^ j;
        if (ixj > i) {
          const float v0 = sv[i], v1 = sv[ixj];
          const unsigned short i0 = si[i], i1 = si[ixj];
          // strict "better" order: larger value first, smaller index on tie
          const bool b1better = (v1 > v0) || (v1 == v0 && i1 < i0);
          const bool desc = ((i & k) == 0);
          if (b1better == desc) {
            sv[i] = v1; sv[ixj] = v0;
            si[i] = i1; si[ixj] = i0;
          }
        }
      }
      __syncthreads();
    }
  }

  for (int i = t; i < MEM_SIZE; i += 1024) {
    const int src = (int)si[i];
    mem_out[b * MEM_SIZE + i]  = mem_input[b * MEM_LEN + src];
    mask_out[b * MEM_SIZE + i] = mask[b * MEM_LEN + src];
  }
}

// ---------------------------------------------------------------------------
extern "C" void kernel_launch(void* const* d_in, const int* in_sizes, int n_in,
                              void* d_out, int out_size, void* d_ws, size_t ws_size,
                              hipStream_t stream) {
  const int*   mem_input = (const int*)d_in[0];
  const int*   mask      = (const int*)d_in[1];
  const float* query     = (const float*)d_in[2];
  const float* emb       = (const float*)d_in[3];
  const float* W_trans   = (const float*)d_in[4];
  const float* W_q       = (const float*)d_in[5];
  const float* W_k       = (const float*)d_in[6];
  // d_in[7] = W_v : dead in the reference (v unused downstream) -> skipped.
  // d_in[8] = mem_size scalar, fixed to 512.

  float* t1     = (float*)d_ws;                 // [32,256]
  float* t2     = t1 + BS * D_MODEL;            // [32,256]
  float* qeff   = t2 + BS * D_MODEL;            // [32,256]
  float* energy = qeff + BS * D_MODEL;          // [32,8192]
  float* attn   = energy + BS * MEM_LEN;        // [32,8192]

  int* mem_out  = (int*)d_out;                  // [32,512]
  int* mask_out = mem_out + BS * MEM_SIZE;      // [32,512]

  // q_eff = query @ W_q @ W_k^T @ W_trans^T   (WMMA f32 16x16x4 chain)
  gemm_qchain<0><<<1, 512, 0, stream>>>(query, W_q,     t1);
  gemm_qchain<1><<<1, 512, 0, stream>>>(t1,    W_k,     t2);
  gemm_qchain<1><<<1, 512, 0, stream>>>(t2,    W_trans, qeff);

  // energy: 262144 gathered 256-dim dots, L2-resident embedding table
  energy_kernel<<<dim3(MEM_LEN / 8, BS), 256, 0, stream>>>(mem_input, qeff, emb, energy);

  // softmax then post-softmax masking
  softmax_mask_kernel<<<BS, 1024, 0, stream>>>(energy, mask, attn);

  // top-512 + gather
  topk_gather_kernel<<<BS, 1024, 0, stream>>>(attn, mem_input, mask, mem_out, mask_out);
}